// MyLinearLayer_1640677507626
// MI455X (gfx1250) — compile-verified
//
#include <hip/hip_runtime.h>

// out[r, c] = x[r, c] * clamp(w[c], 0, 1)   (ROWS=8192, COLS=4096, all f32)
//
// Bandwidth-bound: 268 MB of traffic vs 33.5 MFLOP. Strategy:
//  - float4 (b128) streaming loads/stores with non-temporal hints (don't
//    pollute the 192MB L2 with stream-once data)
//  - weights (16 KiB) staged into LDS per block via gfx1250 async
//    global->LDS copies (ASYNCcnt path), then read conflict-free from LDS
//  - 256-thread blocks (8 waves on wave32), each block = 2 full rows

typedef float f4 __attribute__((ext_vector_type(4)));

#define ROWS 8192
#define COLS 4096
#define COLS4 (COLS / 4)        // 1024 float4 per row
#define BLOCK 256
#define F4_PER_THREAD 8         // 256 * 8 = 2048 f4 per block = 2 rows
#define NBLOCKS ((ROWS * COLS4) / (BLOCK * F4_PER_THREAD))  // 4096

__global__ __launch_bounds__(BLOCK) void col_scale_kernel(
    const f4* __restrict__ x, const float* __restrict__ w, f4* __restrict__ out) {
  __shared__ f4 sw[COLS4];  // 16 KiB clamped-later weight tile
  const int tid = threadIdx.x;

  // ---- Stage the 16 KiB weight vector into LDS with async global->LDS b128
  // copies: 4 issues x 256 lanes x 16B = 16 KiB. The low 32 bits of a flat
  // pointer into LDS are the LDS byte address (ISA 10.2 aperture mapping).
  const f4* wg = (const f4*)w;
#pragma unroll
  for (int i = 0; i < 4; ++i) {
    const f4* src = wg + i * BLOCK + tid;
    unsigned lds_addr = (unsigned)(unsigned long long)(&sw[i * BLOCK + tid]);
    asm volatile("global_load_async_to_lds_b128 %0, %1, off"
                 :
                 : "v"(lds_addr), "v"(src)
                 : "memory");
  }
  asm volatile("s_wait_asynccnt 0" ::: "memory");
  __syncthreads();

  // ---- Stream 2 rows per block. col4 = idx & 1023 since 1024 f4 per row.
  const long long base = (long long)blockIdx.x * (BLOCK * F4_PER_THREAD);
#pragma unroll
  for (int i = 0; i < F4_PER_THREAD; ++i) {
    const long long idx = base + (long long)(i * BLOCK + tid);
    const int col4 = (int)(idx & (long long)(COLS4 - 1));

    f4 v = __builtin_nontemporal_load(&x[idx]);   // TH=NT load
    f4 wc = sw[col4];                             // ds_load_b128, conflict-free

    // clamp(w, 0, 1) at use time (VALU is free at AI = 0.125 flop/byte)
    wc.x = fminf(fmaxf(wc.x, 0.0f), 1.0f);
    wc.y = fminf(fmaxf(wc.y, 0.0f), 1.0f);
    wc.z = fminf(fmaxf(wc.z, 0.0f), 1.0f);
    wc.w = fminf(fmaxf(wc.w, 0.0f), 1.0f);

    v.x *= wc.x;
    v.y *= wc.y;
    v.z *= wc.z;
    v.w *= wc.w;

    __builtin_nontemporal_store(v, &out[idx]);    // TH=NT store
  }
}

extern "C" void kernel_launch(void* const* d_in, const int* in_sizes, int n_in,
                              void* d_out, int out_size, void* d_ws, size_t ws_size,
                              hipStream_t stream) {
  (void)in_sizes; (void)n_in; (void)out_size; (void)d_ws; (void)ws_size;
  const f4* x = (const f4*)d_in[0];          // [ROWS, COLS] f32
  const float* wts = (const float*)d_in[1];  // [COLS] f32
  f4* out = (f4*)d_out;                      // [ROWS, COLS] f32

  col_scale_kernel<<<NBLOCKS, BLOCK, 0, stream>>>(x, wts, out);
}